// BinarizeLayer_5746666242543
// MI455X (gfx1250) — compile-verified
//
#include <hip/hip_runtime.h>
#include <stdint.h>

// BinarizeLayer min-cut DP for MI455X (gfx1250, wave32).
// One lane per row; 32 rows per single-wave workgroup; tiles of 64 columns
// staged through LDS with async global<->LDS DMA (ASYNCcnt path).
// 8-deep buffer ring / 7-tile prefetch: ~56 KB of async traffic in flight per
// wave (~14 MB across 256 waves) to cover HBM latency at 23.3 TB/s.

#define NROWS   8192
#define NCOLS   4096
#define LPEN    512            // int(1024 * 0.5)
#define TILE    64             // columns per tile
#define TPAD    65             // padded LDS row stride (floats) -> bank-conflict-free
#define ROWS    32             // rows per workgroup (== wave32)
#define NTILES  (NCOLS / TILE) // 64
#define CHUNKS  (TILE / 4)     // 16 b128 async ops per tile (each moves 32 lanes x 16B)
#define NBUF    8              // LDS buffer ring
#define DEPTH   7              // prefetch depth (tiles in flight)

// Wait until at most n tiles (n*16 async ops) remain outstanding.
__device__ __forceinline__ void wait_async_tiles(int n) {
    switch (n) {
    case 0:  asm volatile("s_wait_asynccnt 0x0"  ::: "memory"); break;
    case 1:  asm volatile("s_wait_asynccnt 0x10" ::: "memory"); break;
    case 2:  asm volatile("s_wait_asynccnt 0x20" ::: "memory"); break;
    case 3:  asm volatile("s_wait_asynccnt 0x30" ::: "memory"); break;
    case 4:  asm volatile("s_wait_asynccnt 0x40" ::: "memory"); break;
    case 5:  asm volatile("s_wait_asynccnt 0x50" ::: "memory"); break;
    case 6:  asm volatile("s_wait_asynccnt 0x60" ::: "memory"); break;
    default: asm volatile("s_wait_asynccnt 0x70" ::: "memory"); break;
    }
}

// Issue one tile (32 rows x 64 cols) of async loads: global -> LDS.
// Lanes 0..15 of an instruction cover 256 contiguous bytes of row r,
// lanes 16..31 cover row r+1 -> fully coalesced.
__device__ __forceinline__ void tile_load_async(const float* gbase, uint32_t lds, int lane) {
#pragma unroll
    for (int j = 0; j < CHUNKS; ++j) {
        int k = j * 32 + lane;
        int r = k >> 4;            // row within tile
        int c = (k & 15) << 2;     // column (floats)
        uint64_t ga = (uint64_t)(uintptr_t)(gbase + (size_t)r * NCOLS + c);
        uint32_t la = lds + (uint32_t)(r * TPAD + c) * 4u;
        asm volatile("global_load_async_to_lds_b128 %0, %1, off"
                     :: "v"(la), "v"(ga) : "memory");
    }
}

// Issue one tile of async stores: LDS -> global (labels).
__device__ __forceinline__ void tile_store_async(int* gbase, uint32_t lds, int lane) {
#pragma unroll
    for (int j = 0; j < CHUNKS; ++j) {
        int k = j * 32 + lane;
        int r = k >> 4;
        int c = (k & 15) << 2;
        uint64_t ga = (uint64_t)(uintptr_t)(gbase + (size_t)r * NCOLS + c);
        uint32_t la = lds + (uint32_t)(r * TPAD + c) * 4u;
        asm volatile("global_store_async_from_lds_b128 %0, %1, off"
                     :: "v"(ga), "v"(la) : "memory");
    }
}

__global__ __launch_bounds__(32) void binarize_mincut(const float* __restrict__ probs,
                                                      int* __restrict__ out) {
    __shared__ float tile[NBUF][ROWS * TPAD];   // 8 x 8.32 KB = 66.6 KB

    const int lane = threadIdx.x;               // row within block
    const int row0 = blockIdx.x * ROWS;
    const float* gprob = probs + (size_t)row0 * NCOLS;
    int* gout = out + (size_t)row0 * NCOLS;
    // Per-lane choice-bit scratch: low 256 words of this row of d_out.
    // 8 MB total footprint -> resident in the 192 MB L2.
    int* outw = out + (size_t)(row0 + lane) * NCOLS;

    const uint32_t lbase0 = (uint32_t)(uintptr_t)&tile[0][0];
    auto lb = [&](int b) { return lbase0 + (uint32_t)b * (uint32_t)(ROWS * TPAD * 4); };

    // ---------------- forward DP ----------------
#pragma unroll
    for (int t = 0; t < DEPTH; ++t)
        tile_load_async(gprob + (size_t)t * TILE, lb(t), lane);

    int f0 = 0, f1 = 0;
    for (int t = 0; t < NTILES; ++t) {
        if (t + DEPTH < NTILES)
            tile_load_async(gprob + (size_t)(t + DEPTH) * TILE, lb((t + DEPTH) & (NBUF - 1)), lane);
        int ahead = NTILES - 1 - t;
        if (ahead > DEPTH) ahead = DEPTH;
        wait_async_tiles(ahead);                // guarantees tile t has landed
        const float* lrow = &tile[t & (NBUF - 1)][lane * TPAD];
#pragma unroll
        for (int g = 0; g < TILE / 16; ++g) {
            uint32_t cw = 0u;
#pragma unroll
            for (int k = 0; k < 16; ++k) {
                float p = lrow[g * 16 + k];
                int a = (int)floorf(1024.0f * p);
                int b = (int)floorf(1024.0f * (1.0f - p));
                int u0 = f1 + LPEN;
                int u1 = f0 + LPEN;
                int c0 = u0 < f0;
                int c1 = u1 < f1;
                int nf0 = a + (c0 ? u0 : f0);
                int nf1 = b + (c1 ? u1 : f1);
                int m  = nf0 < nf1 ? nf0 : nf1; // renormalize (bounded int32)
                f0 = nf0 - m;
                f1 = nf1 - m;
                cw |= ((uint32_t)c0) << (2 * k);
                cw |= ((uint32_t)c1) << (2 * k + 1);
            }
            outw[t * 4 + g] = (int)cw;          // choice words: L2-resident scratch
        }
    }

    // ---------------- backward traceback ----------------
    int x = (f1 <= f0) ? 1 : 0;                 // label of last node (tie -> 1)
    for (int t = NTILES - 1; t >= 0; --t) {
        // Buffer t&7 was last used by tile t+8's stores; allow up to 7 newer
        // tiles of stores to remain in flight.
        wait_async_tiles(DEPTH);
        uint32_t w[4];
#pragma unroll
        for (int g = 0; g < 4; ++g) w[g] = (uint32_t)outw[t * 4 + g];
        int* lrow = (int*)&tile[t & (NBUF - 1)][lane * TPAD];
#pragma unroll
        for (int g = 3; g >= 0; --g) {
            uint32_t cw = w[g];
#pragma unroll
            for (int k = 15; k >= 0; --k) {
                lrow[g * 16 + k] = x;           // label at position t*64+g*16+k
                int c0 = (int)((cw >> (2 * k)) & 1u);
                int c1 = (int)((cw >> (2 * k + 1)) & 1u);
                x = x ? (c1 ^ 1) : c0;          // x_{i-1} = x_i ? !c1[i] : c0[i]
            }
        }
        asm volatile("s_wait_dscnt 0x0" ::: "memory");  // LDS writes visible to DMA
        tile_store_async(gout + (size_t)t * TILE, lb(t & (NBUF - 1)), lane);
    }
    wait_async_tiles(0);
}

extern "C" void kernel_launch(void* const* d_in, const int* in_sizes, int n_in,
                              void* d_out, int out_size, void* d_ws, size_t ws_size,
                              hipStream_t stream) {
    (void)in_sizes; (void)n_in; (void)out_size; (void)d_ws; (void)ws_size;
    const float* probs = (const float*)d_in[0];
    int* out = (int*)d_out;
    dim3 grid(NROWS / ROWS);
    dim3 block(32);
    hipLaunchKernelGGL(binarize_mincut, grid, block, 0, stream, probs, out);
}